// PairEmbed_17918603559412
// MI455X (gfx1250) — compile-verified
//
#include <hip/hip_runtime.h>
#include <math.h>

typedef __attribute__((ext_vector_type(2))) float v2f;
typedef __attribute__((ext_vector_type(8))) float v8f;

#define BATCH 64
#define S 128
#define P_TOT 8256      // S*(S+1)/2
#define NT 64           // pairs per workgroup tile
#define CH 64           // channels
#define NTILES 129      // P_TOT / NT (exact)
#define PI_F     3.14159265358979323846f
#define TWO_PI_F 6.28318530717958647692f
#define EPS_F    1e-8f
#define BN_EPS_F 1e-5f

__device__ __forceinline__ float gelu_exact(float x) {
    return 0.5f * x * (1.0f + erff(x * 0.70710678118654752f));
}

__launch_bounds__(256)
__global__ void pairembed_fused(
    const float* __restrict__ x,
    const float* __restrict__ W0, const float* __restrict__ b0,
    const float* __restrict__ g0, const float* __restrict__ be0,
    const float* __restrict__ rm0, const float* __restrict__ rv0,
    const float* __restrict__ W1, const float* __restrict__ b1,
    const float* __restrict__ g1, const float* __restrict__ be1,
    const float* __restrict__ rm1, const float* __restrict__ rv1,
    const float* __restrict__ W2, const float* __restrict__ b2,
    const float* __restrict__ g2, const float* __restrict__ be2,
    const float* __restrict__ rm2, const float* __restrict__ rv2,
    float* __restrict__ out)
{
    __shared__ float sW1[CH * CH];     // 16 KB
    __shared__ float sW2[CH * CH];     // 16 KB
    __shared__ float sHA[CH * NT];     // 16 KB  (layer activations, ping)
    __shared__ float sHB[CH * NT];     // 16 KB  (pong)
    __shared__ float sW0[CH * 3];
    __shared__ float sSc[3][CH];
    __shared__ float sSh[3][CH];
    __shared__ float sF[3][NT];
    __shared__ int   sI[NT];
    __shared__ int   sJ[NT];

    const int t     = threadIdx.x;
    const int wave  = t >> 5;
    const int lane  = t & 31;
    const int b     = blockIdx.y;
    const int ptile = blockIdx.x;

    // ---- stage weights W1, W2 (float4 loads) ----
    {
        const float4* w1v = (const float4*)W1;
        const float4* w2v = (const float4*)W2;
        float4* s1 = (float4*)sW1;
        float4* s2 = (float4*)sW2;
        #pragma unroll
        for (int k = 0; k < (CH * CH / 4) / 256; ++k) {
            int idx = t + k * 256;
            s1[idx] = w1v[idx];
            s2[idx] = w2v[idx];
        }
    }
    if (t < CH * 3) sW0[t] = W0[t];

    // ---- folded BN scale/shift per layer: y = acc*sc + sh ----
    if (t < 3 * CH) {
        int l = t >> 6, ch = t & (CH - 1);
        const float *bb, *gg, *bee, *rmm, *rvv;
        if (l == 0)      { bb = b0; gg = g0; bee = be0; rmm = rm0; rvv = rv0; }
        else if (l == 1) { bb = b1; gg = g1; bee = be1; rmm = rm1; rvv = rv1; }
        else             { bb = b2; gg = g2; bee = be2; rmm = rm2; rvv = rv2; }
        float sc = gg[ch] / sqrtf(rvv[ch] + BN_EPS_F);
        sSc[l][ch] = sc;
        sSh[l][ch] = (bb[ch] - rmm[ch]) * sc + bee[ch];
    }

    // ---- pairwise features for this tile of 64 pairs ----
    if (t < NT) {
        int p = ptile * NT + t;
        int i = (int)floorf((sqrtf(8.0f * (float)p + 1.0f) - 1.0f) * 0.5f);
        while ((i + 1) * (i + 2) / 2 <= p) ++i;
        while (i * (i + 1) / 2 > p) --i;
        int j = p - i * (i + 1) / 2;
        sI[t] = i; sJ[t] = j;
        const float* xb = x + (size_t)b * 3 * S;
        float pti = xb[0 * S + i], rapi = xb[1 * S + i], phii = xb[2 * S + i];
        float ptj = xb[0 * S + j], rapj = xb[1 * S + j], phij = xb[2 * S + j];
        float d = phii - phij + PI_F;
        float m = fmodf(d, TWO_PI_F);
        if (m < 0.0f) m += TWO_PI_F;
        float dphi = m - PI_F;
        float drap = rapi - rapj;
        float delta = sqrtf(drap * drap + dphi * dphi);
        float lndelta = logf(fmaxf(delta, EPS_F));
        float ptmin = fminf(pti, ptj);
        float lnkt = logf(fmaxf(ptmin * delta, EPS_F));
        float lnz  = logf(fmaxf(ptmin / fmaxf(pti + ptj, EPS_F), EPS_F));
        sF[0][t] = lnkt; sF[1][t] = lnz; sF[2][t] = lndelta;
    }
    __syncthreads();

    // ---- layer 0: 3 -> 64, scalar FMAs (K=3), BN + GELU -> sHA[ch][n] ----
    {
        int n  = t & (NT - 1);
        int ob = (t >> 6) * 16;
        float f0 = sF[0][n], f1 = sF[1][n], f2 = sF[2][n];
        #pragma unroll
        for (int k = 0; k < 16; ++k) {
            int o = ob + k;
            float h = sW0[o * 3 + 0] * f0 + sW0[o * 3 + 1] * f1 + sW0[o * 3 + 2] * f2;
            h = h * sSc[0][o] + sSh[0][o];
            sHA[o * NT + n] = gelu_exact(h);
        }
    }
    __syncthreads();

    // ---- WMMA GEMM layer: dst = postproc(Wl * src), 64x64x64 in fp32 ----
    // A = weights 16x4 tiles, B = activations 4x16 tiles, per documented layouts.
    auto wmma_layer = [&](const float* sWl, const float* src, float* dst,
                          int li, bool do_gelu) {
        const int half  = lane >> 4;          // 0: lanes 0-15, 1: lanes 16-31
        const int l15   = lane & 15;
        const int khalf = half << 1;          // K sub-offset 0 or 2
        #pragma unroll
        for (int tt = 0; tt < 2; ++tt) {
            int tile = wave + tt * 8;         // 0..15
            int mt = tile >> 2, nt = tile & 3;
            int row = mt * 16 + l15;          // A matrix row (out channel)
            int col = nt * 16 + l15;          // B matrix col (pair)
            v8f acc = {};
            #pragma unroll
            for (int kt = 0; kt < 16; ++kt) {
                int kb = kt * 4 + khalf;
                v2f a, bv;
                a[0]  = sWl[row * CH + kb];
                a[1]  = sWl[row * CH + kb + 1];
                bv[0] = src[kb * NT + col];
                bv[1] = src[(kb + 1) * NT + col];
                acc = __builtin_amdgcn_wmma_f32_16x16x4_f32(
                          false, a, false, bv, (short)0, acc, false, false);
            }
            // C/D layout: VGPR v -> M = v (lanes 0-15) / v+8 (lanes 16-31)
            int chbase = mt * 16 + (half << 3);
            #pragma unroll
            for (int v = 0; v < 8; ++v) {
                int ch = chbase + v;
                float h = acc[v] * sSc[li][ch] + sSh[li][ch];
                if (do_gelu) h = gelu_exact(h);
                dst[ch * NT + col] = h;
            }
        }
    };

    wmma_layer(sW1, sHA, sHB, 1, true);   // layer 1 + GELU
    __syncthreads();
    wmma_layer(sW2, sHB, sHA, 2, false);  // layer 2 (no GELU)
    __syncthreads();

    // ---- scatter lower triangle: y[b, o, i, j] (coalesced along j) ----
    {
        int n  = t & (NT - 1);
        int ob = (t >> 6) * 16;
        int i = sI[n], j = sJ[n];
        float* outb = out + (size_t)b * CH * S * S;
        #pragma unroll
        for (int k = 0; k < 16; ++k) {
            int o = ob + k;
            outb[((size_t)o * S + i) * S + j] = sHA[o * NT + n];
        }
    }
}

// Mirror the lower triangle to the strict upper triangle via LDS transpose.
// grid.x: 10 tile-pairs (ti >= tj) of 32x32 tiles; grid.y: B*CH planes.
__launch_bounds__(256)
__global__ void sym_mirror(float* __restrict__ y)
{
    __shared__ float tile[32][33];
    const int ti_tab[10] = {0,1,1,2,2,2,3,3,3,3};
    const int tj_tab[10] = {0,0,1,0,1,2,0,1,2,3};
    int ti = ti_tab[blockIdx.x], tj = tj_tab[blockIdx.x];
    size_t plane = (size_t)blockIdx.y * (S * S);
    int lx = threadIdx.x & 31;
    int ly = threadIdx.x >> 5;

    #pragma unroll
    for (int r0 = 0; r0 < 32; r0 += 8) {
        int r = r0 + ly;
        tile[r][lx] = y[plane + (size_t)(ti * 32 + r) * S + (tj * 32 + lx)];
    }
    __syncthreads();
    #pragma unroll
    for (int r0 = 0; r0 < 32; r0 += 8) {
        int r  = r0 + ly;
        int gi = tj * 32 + r;      // upper-triangle row
        int gj = ti * 32 + lx;     // upper-triangle col
        if (gj > gi)
            y[plane + (size_t)gi * S + gj] = tile[lx][r];
    }
}

extern "C" void kernel_launch(void* const* d_in, const int* in_sizes, int n_in,
                              void* d_out, int out_size, void* d_ws, size_t ws_size,
                              hipStream_t stream) {
    (void)in_sizes; (void)n_in; (void)out_size; (void)d_ws; (void)ws_size;
    const float* x   = (const float*)d_in[0];
    const float* W0  = (const float*)d_in[1];
    const float* b0  = (const float*)d_in[2];
    const float* g0  = (const float*)d_in[3];
    const float* be0 = (const float*)d_in[4];
    const float* rm0 = (const float*)d_in[5];
    const float* rv0 = (const float*)d_in[6];
    const float* W1  = (const float*)d_in[7];
    const float* b1  = (const float*)d_in[8];
    const float* g1  = (const float*)d_in[9];
    const float* be1 = (const float*)d_in[10];
    const float* rm1 = (const float*)d_in[11];
    const float* rv1 = (const float*)d_in[12];
    const float* W2  = (const float*)d_in[13];
    const float* b2  = (const float*)d_in[14];
    const float* g2  = (const float*)d_in[15];
    const float* be2 = (const float*)d_in[16];
    const float* rm2 = (const float*)d_in[17];
    const float* rv2 = (const float*)d_in[18];
    float* out = (float*)d_out;

    dim3 grid1(NTILES, BATCH);
    pairembed_fused<<<grid1, 256, 0, stream>>>(
        x, W0, b0, g0, be0, rm0, rv0,
           W1, b1, g1, be1, rm1, rv1,
           W2, b2, g2, be2, rm2, rv2, out);

    dim3 grid2(10, BATCH * CH);
    sym_mirror<<<grid2, 256, 0, stream>>>(out);
}